// RotatedROIPool_83416854823436
// MI455X (gfx1250) — compile-verified
//
#include <hip/hip_runtime.h>
#include <hip/hip_bf16.h>

typedef float v2f __attribute__((ext_vector_type(2)));
typedef float v4f __attribute__((ext_vector_type(4)));
typedef float v8f __attribute__((ext_vector_type(8)));

// Problem constants (from the reference setup)
#define SS 64          // crop size S
#define FEAT_DOWN 8
#define EXPANDF 1.25f
#define BB 2
#define NB 64          // boxes per image
#define CC 256
#define HF 128
#define WF 128
#define CH_STRIDE (HF*WF)     // 16384 floats per channel plane
#define CSPLIT 4              // channel partitions per ROI
#define CPB (CC/CSPLIT)       // 64 channels per block

__device__ __forceinline__ void obb_params(const float* __restrict__ obb, int roi,
                                           float& co, float& si,
                                           float& cxf, float& cyf,
                                           float& wf, float& hf)
{
    const float cx  = obb[roi*5 + 0];
    const float cy  = obb[roi*5 + 1];
    const float w   = obb[roi*5 + 2];
    const float h   = obb[roi*5 + 3];
    const float deg = obb[roi*5 + 4];
    cxf = cx * (1.0f / FEAT_DOWN);
    cyf = cy * (1.0f / FEAT_DOWN);
    wf  = fmaxf(w, 1.0f) * (EXPANDF / FEAT_DOWN);
    hf  = fmaxf(h, 1.0f) * (EXPANDF / FEAT_DOWN);
    const float ang = deg * 0.017453292519943295f;  // deg2rad
    si = __sinf(ang);
    co = __cosf(ang);
}

// ---------------------------------------------------------------------------
// Crops kernel (unchanged, roofline-shaped): one block = one ROI x 16 rows x
// 64 channels; each thread owns 4 consecutive sx -> 16B NT b128 stores
// (512B/wave/instr into the dominant 537MB write stream), 32 in-flight b32
// gathers from the L2-resident feature map, prefetch 8 planes ahead.
// ---------------------------------------------------------------------------
__global__ __launch_bounds__(256) void rroi_crops_kernel(
    const float* __restrict__ feat,
    const float* __restrict__ obb,
    float* __restrict__ out)
{
    const int roi = blockIdx.x;                         // uniform -> scalar obb loads
    const int c0  = blockIdx.z * CPB;
    const int sy  = (blockIdx.y << 4) + (threadIdx.x >> 4);   // 16 rows / block
    const int sx0 = (threadIdx.x & 15) << 2;                  // 4 consecutive sx
    const int b   = roi >> 6;                                 // roi / NB

    float co, si, cxf, cyf, wf, hf;
    obb_params(obb, roi, co, si, cxf, cyf, wf, hf);

    const float t00 =  co * (wf * (2.0f / WF));
    const float t01 = -si * (hf * (2.0f / HF));
    const float t02 =  cxf * (2.0f / WF) - 1.0f;
    const float t10 =  si * (wf * (2.0f / WF));
    const float t11 =  co * (hf * (2.0f / HF));
    const float t12 =  cyf * (2.0f / HF) - 1.0f;

    const float ys = (2.0f * (float)sy + 1.0f) * (1.0f / SS) - 1.0f;

    int   o[16];
    float w[16];
#pragma unroll
    for (int j = 0; j < 4; ++j) {
        const float xs = (2.0f * (float)(sx0 + j) + 1.0f) * (1.0f / SS) - 1.0f;
        const float gx = t00 * xs + t01 * ys + t02;
        const float gy = t10 * xs + t11 * ys + t12;
        const float ix = ((gx + 1.0f) * (float)WF - 1.0f) * 0.5f;
        const float iy = ((gy + 1.0f) * (float)HF - 1.0f) * 0.5f;
        const float fx0 = floorf(ix);
        const float fy0 = floorf(iy);
        const int   x0  = (int)fx0,  y0 = (int)fy0;
        const int   x1  = x0 + 1,    y1 = y0 + 1;
        const float wx1 = ix - fx0,  wy1 = iy - fy0;
        const float wx0 = 1.0f - wx1, wy0 = 1.0f - wy1;
        const float vx0 = (x0 >= 0 && x0 < WF) ? 1.0f : 0.0f;
        const float vx1 = (x1 >= 0 && x1 < WF) ? 1.0f : 0.0f;
        const float vy0 = (y0 >= 0 && y0 < HF) ? 1.0f : 0.0f;
        const float vy1 = (y1 >= 0 && y1 < HF) ? 1.0f : 0.0f;
        const int x0c = min(max(x0, 0), WF - 1);
        const int x1c = min(max(x1, 0), WF - 1);
        const int y0c = min(max(y0, 0), HF - 1);
        const int y1c = min(max(y1, 0), HF - 1);
        o[j*4 + 0] = y0c * WF + x0c;  w[j*4 + 0] = wx0 * wy0 * vx0 * vy0;
        o[j*4 + 1] = y0c * WF + x1c;  w[j*4 + 1] = wx1 * wy0 * vx1 * vy0;
        o[j*4 + 2] = y1c * WF + x0c;  w[j*4 + 2] = wx0 * wy1 * vx0 * vy1;
        o[j*4 + 3] = y1c * WF + x1c;  w[j*4 + 3] = wx1 * wy1 * vx1 * vy1;
    }

    const float* __restrict__ fb = feat + ((size_t)b * CC + c0) * CH_STRIDE;
    const float* q[16];
#pragma unroll
    for (int k = 0; k < 16; ++k) q[k] = fb + o[k];

    float* __restrict__ op = out + (size_t)roi * CC * (SS*SS)
                                 + (size_t)c0 * (SS*SS) + sy * SS + sx0;

    for (int cc = 0; cc < CPB; cc += 2) {
        const float* pf0 = (cc + 8 < CPB) ? (q[0] + 8 * CH_STRIDE) : q[0];
        const float* pf1 = (cc + 8 < CPB) ? (q[8] + 8 * CH_STRIDE) : q[8];
        __builtin_prefetch(pf0, 0, 3);
        __builtin_prefetch(pf1, 0, 3);
#pragma unroll
        for (int u = 0; u < 2; ++u) {
            v4f v;
#pragma unroll
            for (int j = 0; j < 4; ++j) {
                const float a  = q[j*4 + 0][u * CH_STRIDE];
                const float bv = q[j*4 + 1][u * CH_STRIDE];
                const float c2 = q[j*4 + 2][u * CH_STRIDE];
                const float d2 = q[j*4 + 3][u * CH_STRIDE];
                v[j] = fmaf(w[j*4 + 0], a,
                       fmaf(w[j*4 + 1], bv,
                       fmaf(w[j*4 + 2], c2,
                            w[j*4 + 3] * d2)));
            }
            __builtin_nontemporal_store(v, (v4f*)(op + u * (SS*SS)));  // b128 NT
        }
#pragma unroll
        for (int k = 0; k < 16; ++k) q[k] += 2 * CH_STRIDE;
        op += 2 * (SS*SS);
    }
}

// ---------------------------------------------------------------------------
// M matrices: [B, N, 2, 3]. M[:, :2] = R(theta) @ diag(wf/S, hf/S) is a true
// matrix product -> map 8 ROIs per wave onto one V_WMMA_F32_16X16X4_F32 via
// block-diagonal embedding (shared K slots 0..1; off-diagonal blocks unused).
// fp32 WMMA products here are bit-identical to the scalar formula.
// 512 threads = 16 wave32 waves = 128 ROIs, one WMMA per wave.
// ---------------------------------------------------------------------------
#if defined(__gfx1250__) && __has_builtin(__builtin_amdgcn_wmma_f32_16x16x4_f32)
#define RROI_HAVE_WMMA 1
#endif

__global__ __launch_bounds__(512) void rroi_M_kernel(const float* __restrict__ obb,
                                                     float* __restrict__ mout)
{
#if defined(RROI_HAVE_WMMA)
    __shared__ float Dm[16][256];                 // [wave][m*16+n], 16KB

    const int wv      = threadIdx.x >> 5;         // wave 0..15
    const int lane    = threadIdx.x & 31;
    const int roiBase = wv * 8;

    // A (16x4 f32, 2 VGPRs): lanes 0-15 hold rows M=lane with K=0,1;
    // lanes 16-31 hold K=2,3 -> zeroed (kills cross terms).
    // B (4x16 f32, 2 VGPRs): lanes 0-15 hold cols N=lane with K=0,1 rows.
    v2f a = {0.0f, 0.0f};
    v2f bm = {0.0f, 0.0f};
    if (lane < 16) {
        const int g   = lane >> 1;                // ROI within wave
        const int rc  = lane & 1;                 // row (for A) / col (for B)
        float co, si, cxf, cyf, wf, hf;
        obb_params(obb, roiBase + g, co, si, cxf, cyf, wf, hf);
        // A[m, 0..1] = R[r][0..1],  R = [[cos, -sin],[sin, cos]]
        a.x = (rc == 0) ?  co : si;
        a.y = (rc == 0) ? -si : co;
        // B[0..1, n] = diag(wf/S, hf/S)[0..1][c]
        bm.x = (rc == 0) ? wf * (1.0f / SS) : 0.0f;
        bm.y = (rc == 1) ? hf * (1.0f / SS) : 0.0f;
    }

    v8f acc = {};
    // 8 args: (neg_a, A, neg_b, B, c_mod, C, reuse_a, reuse_b)
    acc = __builtin_amdgcn_wmma_f32_16x16x4_f32(false, a, false, bm,
                                                (short)0, acc, false, false);

    // D layout: VGPR i holds M=i (lanes 0-15) / M=i+8 (lanes 16-31), N=lane&15.
    {
        const int n     = lane & 15;
        const int mbase = (lane >= 16) ? 8 : 0;
#pragma unroll
        for (int i = 0; i < 8; ++i)
            Dm[wv][(i + mbase) * 16 + n] = acc[i];
    }
    __syncthreads();

    if (lane < 8) {
        const int g   = lane;
        const int roi = roiBase + g;
        const float A_ = Dm[wv][(2*g + 0) * 16 + (2*g + 0)];
        const float Bv = Dm[wv][(2*g + 0) * 16 + (2*g + 1)];
        const float A2 = Dm[wv][(2*g + 1) * 16 + (2*g + 0)];
        const float B2 = Dm[wv][(2*g + 1) * 16 + (2*g + 1)];
        float co, si, cxf, cyf, wf, hf;
        obb_params(obb, roi, co, si, cxf, cyf, wf, hf);
        const float Cx = cxf - 0.5f * (float)SS * (A_ + Bv);
        const float Cy = cyf - 0.5f * (float)SS * (A2 + B2);
        float* p = mout + roi * 6;
        p[0] = A_; p[1] = Bv; p[2] = Cx;
        p[3] = A2; p[4] = B2; p[5] = Cy;
    }
#else
    // Scalar fallback (also the host-pass body), same launch geometry.
    const int idx = threadIdx.x;
    if (idx < BB * NB) {
        float co, si, cxf, cyf, wf, hf;
        obb_params(obb, idx, co, si, cxf, cyf, wf, hf);
        const float A  =  co * (wf * (1.0f / SS));
        const float Bv = -si * (hf * (1.0f / SS));
        const float A2 =  si * (wf * (1.0f / SS));
        const float B2 =  co * (hf * (1.0f / SS));
        const float Cx = cxf - 0.5f * (float)SS * (A + Bv);
        const float Cy = cyf - 0.5f * (float)SS * (A2 + B2);
        float* p = mout + idx * 6;
        p[0] = A;  p[1] = Bv; p[2] = Cx;
        p[3] = A2; p[4] = B2; p[5] = Cy;
    }
#endif
}

extern "C" void kernel_launch(void* const* d_in, const int* in_sizes, int n_in,
                              void* d_out, int out_size, void* d_ws, size_t ws_size,
                              hipStream_t stream)
{
    const float* feat = (const float*)d_in[0];   // [2,256,128,128] fp32
    const float* obb  = (const float*)d_in[1];   // [2,64,5] fp32
    float* out  = (float*)d_out;                 // crops then M, fp32
    float* mout = out + (size_t)BB * NB * CC * (SS*SS);

    dim3 grid(BB * NB, SS / 16, CSPLIT);         // 128 ROIs x 4 row-groups x 4 ch-splits
    rroi_crops_kernel<<<grid, 256, 0, stream>>>(feat, obb, out);
    rroi_M_kernel<<<1, 512, 0, stream>>>(obb, mout);
}